// SimilarityAwareChoiceModelImproved_60129542144354
// MI455X (gfx1250) — compile-verified
//
#include <hip/hip_runtime.h>
#include <hip/hip_bf16.h>
#include <math.h>

#define NTOK   2048
#define BATCH  4
#define HEADS  4
#define DHEAD  16
#define DMODEL 64
#define NPF    31

typedef float v2f __attribute__((ext_vector_type(2)));
typedef float v8f __attribute__((ext_vector_type(8)));

__device__ __forceinline__ float sp_(float x) {
    return x > 20.f ? x : log1pf(__expf(x));
}

// ---------------------------------------------------------------------------
// Kernel 1: per-row MLPs + feature projection + Q/K/V.  K stored transposed
// per (b,h) as (DH, N) so attention B-operand loads are coalesced.
// ---------------------------------------------------------------------------
__global__ __launch_bounds__(256) void prep_kernel(
    const float* __restrict__ X,
    const float* __restrict__ bw1, const float* __restrict__ bb1,
    const float* __restrict__ bw2, const float* __restrict__ bb2,
    const float* __restrict__ bw3, const float* __restrict__ bb3,
    const float* __restrict__ w_raw, const float* __restrict__ a_raw,
    const float* __restrict__ cs,  const float* __restrict__ p_bias,
    const float* __restrict__ fpw, const float* __restrict__ fpb,
    const float* __restrict__ qw,  const float* __restrict__ qb,
    const float* __restrict__ kw,  const float* __restrict__ kb,
    const float* __restrict__ vw,  const float* __restrict__ vb,
    float* __restrict__ Q, float* __restrict__ Kt, float* __restrict__ V,
    float* __restrict__ utilBase)
{
    __shared__ float npx[4][NPF];
    __shared__ float h1s[4][DMODEL];
    __shared__ float h2s[4][DMODEL];
    __shared__ float zfs[4][DMODEL];

    const int r   = threadIdx.x >> 6;   // row within workgroup (0..3)
    const int col = threadIdx.x & 63;   // output column (0..63)
    const int row = blockIdx.x * 4 + r; // 0 .. B*N-1
    const int b   = row >> 11;
    const int n   = row & (NTOK - 1);
    const float* xr = X + (size_t)row * 32;

    if (col < NPF) npx[r][col] = xr[1 + col];
    __syncthreads();

    float a1 = bb1[col];
    float zf = fpb[col];
    for (int i = 0; i < NPF; ++i) {
        const float xv = npx[r][i];
        a1 += xv * bw1[i * DMODEL + col];
        zf += xv * fpw[i * DMODEL + col];
    }
    h1s[r][col] = fmaxf(a1, 0.f);
    zfs[r][col] = zf;
    __syncthreads();

    float a2 = bb2[col];
    float q = qb[col], k = kb[col], v = vb[col];
    for (int i = 0; i < DMODEL; ++i) {
        a2 += h1s[r][i] * bw2[i * DMODEL + col];
        const float zz = zfs[r][i];
        q += zz * qw[i * DMODEL + col];
        k += zz * kw[i * DMODEL + col];
        v += zz * vw[i * DMODEL + col];
    }
    h2s[r][col] = fmaxf(a2, 0.f);

    const int h = col >> 4, d = col & 15;
    const size_t bh = (size_t)(b * HEADS + h);
    Q [(bh * NTOK + n) * DHEAD + d] = q;
    Kt[(bh * DHEAD + d) * NTOK + n] = k;
    V [(bh * NTOK + n) * DHEAD + d] = v;
    __syncthreads();

    if (col == 0) {
        float bu = bb3[0];
        for (int i = 0; i < DMODEL; ++i) bu += h2s[r][i] * bw3[i];
        const float price = xr[0];
        float pu = p_bias[0];
        for (int kk = 0; kk < 8; ++kk)
            pu -= sp_(sp_(a_raw[kk]) * (price - cs[kk])) * sp_(w_raw[kk]);
        utilBase[row] = bu + pu;
    }
}

// ---------------------------------------------------------------------------
// Kernel 2: top-16 (diag-excluded) per sim row -> dense bias matrix
// biasD[i][j] = sp(tau) * sim_masked / clip(rowsum, 1e-8).
// One wave32 per row; picked set tracked as a per-lane 64-bit bitmask over
// the 64 strided elements each lane owns (no register arrays -> no spills).
// ---------------------------------------------------------------------------
__global__ __launch_bounds__(256) void topk_kernel(
    const float* __restrict__ sim,
    const float* __restrict__ tau_raw,
    float* __restrict__ biasD)
{
    const int lane = threadIdx.x & 31;
    const int row  = blockIdx.x * 8 + (threadIdx.x >> 5);
    const float* sr = sim + (size_t)row * NTOK;

    unsigned long long mask = 0ull;
    float sum = 0.f;
    for (int t = 0; t < 16; ++t) {
        float best = -1.f;
        int   bidx = 0x7fffffff;
        for (int j = lane; j < NTOK; j += 32) {
            if (j == row) continue;
            if ((mask >> (j >> 5)) & 1ull) continue;
            const float vv = fmaxf(sr[j], 0.f);
            if (vv > best || (vv == best && j < bidx)) { best = vv; bidx = j; }
        }
#pragma unroll
        for (int m = 1; m < 32; m <<= 1) {
            const float ov = __shfl_xor(best, m, 32);
            const int   oi = __shfl_xor(bidx, m, 32);
            if (ov > best || (ov == best && oi < bidx)) { best = ov; bidx = oi; }
        }
        sum += best;
        if ((bidx & 31) == lane) mask |= (1ull << (bidx >> 5));
    }

    const float scale = sp_(tau_raw[0]) / fmaxf(sum, 1e-8f);
    for (int j = lane; j < NTOK; j += 32) {
        float vv = 0.f;
        if ((mask >> (j >> 5)) & 1ull) vv = fmaxf(sr[j], 0.f) * scale;
        biasD[(size_t)row * NTOK + j] = vv;
    }
}

// ---------------------------------------------------------------------------
// Kernel 3: flash attention.  One wave per (b,h, 16-query tile).
// S tile (16x16) = sum_c  wmma_f32_16x16x4( Q_chunk, Kt_chunk, S )
// online softmax (16-lane butterflies per half), P relaid out via LDS,
// O tile += sum_c  wmma_f32_16x16x4( P_chunk, V_chunk, O ).
// ---------------------------------------------------------------------------
__global__ __launch_bounds__(256) void flash_kernel(
    const float* __restrict__ Q,
    const float* __restrict__ Kt,
    const float* __restrict__ V,
    const float* __restrict__ biasD,
    float* __restrict__ AO)
{
    __shared__ float plds[8 * 256];        // 256 floats per wave (P tile)
    const int lane = threadIdx.x & 31;
    const int wv   = threadIdx.x >> 5;
    const int gw   = blockIdx.x * 8 + wv;  // 0 .. 2047
    const int bh   = gw >> 7;              // 0 .. 15
    const int qt   = (gw & 127) << 4;      // query tile base
    const int lm   = lane & 15;
    const int lh   = lane >> 4;
    float* P = &plds[wv * 256];

    const float* Qb  = Q  + (size_t)bh * NTOK * DHEAD;
    const float* Ktb = Kt + (size_t)bh * DHEAD * NTOK;
    const float* Vb  = V  + (size_t)bh * NTOK * DHEAD;

    // Q in A-layout: lane m holds row (qt+lm); VGPR pair per K-chunk,
    // half-wave selects K and K+2 (ISA 7.12.2, 32-bit A 16x4).
    v2f aq[4];
#pragma unroll
    for (int c = 0; c < 4; ++c) {
        const int d = 4 * c + 2 * lh;
        aq[c][0] = Qb[(qt + lm) * DHEAD + d];
        aq[c][1] = Qb[(qt + lm) * DHEAD + d + 1];
    }

    v8f o = {0.f, 0.f, 0.f, 0.f, 0.f, 0.f, 0.f, 0.f};
    float rm[8], rl[8];
#pragma unroll
    for (int r = 0; r < 8; ++r) { rm[r] = -__builtin_inff(); rl[r] = 0.f; }

    for (int jt = 0; jt < NTOK; jt += 16) {
        if (jt + 16 < NTOK) {
            __builtin_prefetch(&Vb[(jt + 16) * DHEAD], 0, 1);
            __builtin_prefetch(&Ktb[jt + 16], 0, 1);
        }
        // ---- S = Q * K^T  (4 chained f32 WMMAs, K-dim 16 = 4x4) ----
        v8f s = {0.f, 0.f, 0.f, 0.f, 0.f, 0.f, 0.f, 0.f};
#pragma unroll
        for (int c = 0; c < 4; ++c) {
            const int d = 4 * c + 2 * lh;
            v2f bk;
            bk[0] = Ktb[d * NTOK + jt + lm];
            bk[1] = Ktb[(d + 1) * NTOK + jt + lm];
            s = __builtin_amdgcn_wmma_f32_16x16x4_f32(
                    false, aq[c], false, bk, (short)0, s, false, false);
        }
        // ---- scale + dense similarity bias (C-layout coalesced load) ----
#pragma unroll
        for (int r = 0; r < 8; ++r) {
            s[r] = s[r] * 0.25f +
                   biasD[(size_t)(qt + r + 8 * lh) * NTOK + jt + lm];
        }
        // ---- online softmax: per C-row reductions over 16-lane groups ----
#pragma unroll
        for (int r = 0; r < 8; ++r) {
            float mx = s[r];
            mx = fmaxf(mx, __shfl_xor(mx, 1, 32));
            mx = fmaxf(mx, __shfl_xor(mx, 2, 32));
            mx = fmaxf(mx, __shfl_xor(mx, 4, 32));
            mx = fmaxf(mx, __shfl_xor(mx, 8, 32));
            const float nm   = fmaxf(rm[r], mx);
            const float corr = __expf(rm[r] - nm);
            const float p    = __expf(s[r] - nm);
            float ss = p;
            ss += __shfl_xor(ss, 1, 32);
            ss += __shfl_xor(ss, 2, 32);
            ss += __shfl_xor(ss, 4, 32);
            ss += __shfl_xor(ss, 8, 32);
            rl[r] = rl[r] * corr + ss;
            rm[r] = nm;
            o[r] *= corr;
            s[r]  = p;
        }
        // ---- relayout P: C-layout -> LDS (row-major 16x16) -> A-layout ----
#pragma unroll
        for (int r = 0; r < 8; ++r)
            P[(r + 8 * lh) * 16 + lm] = s[r];

#pragma unroll
        for (int c = 0; c < 4; ++c) {
            const int kc = 4 * c + 2 * lh;
            v2f ap, bv;
            ap[0] = P[lm * 16 + kc];
            ap[1] = P[lm * 16 + kc + 1];
            bv[0] = Vb[(jt + kc) * DHEAD + lm];
            bv[1] = Vb[(jt + kc + 1) * DHEAD + lm];
            o = __builtin_amdgcn_wmma_f32_16x16x4_f32(
                    false, ap, false, bv, (short)0, o, false, false);
        }
    }

#pragma unroll
    for (int r = 0; r < 8; ++r)
        AO[((size_t)bh * NTOK + qt + r + 8 * lh) * DHEAD + lm] = o[r] / rl[r];
}

// ---------------------------------------------------------------------------
// Kernel 4: output projection + interaction MLP -> final per-item utility.
// ---------------------------------------------------------------------------
__global__ __launch_bounds__(256) void post_kernel(
    const float* __restrict__ AO,
    const float* __restrict__ ow,  const float* __restrict__ ob,
    const float* __restrict__ iw1, const float* __restrict__ ib1,
    const float* __restrict__ iw2, const float* __restrict__ ib2,
    const float* __restrict__ utilBase,
    const float* __restrict__ attn_scale,
    float* __restrict__ utilFinal)
{
    __shared__ float av[4][DMODEL];
    __shared__ float aos[4][DMODEL];
    __shared__ float hh[4][DMODEL];

    const int r   = threadIdx.x >> 6;
    const int col = threadIdx.x & 63;
    const int row = blockIdx.x * 4 + r;
    const int b   = row >> 11;
    const int n   = row & (NTOK - 1);
    const int h   = col >> 4, d = col & 15;

    av[r][col] = AO[((size_t)(b * HEADS + h) * NTOK + n) * DHEAD + d];
    __syncthreads();

    float acc = ob[col];
    for (int i = 0; i < DMODEL; ++i) acc += av[r][i] * ow[i * DMODEL + col];
    aos[r][col] = acc;
    __syncthreads();

    float a2 = ib1[col];
    for (int i = 0; i < DMODEL; ++i) a2 += aos[r][i] * iw1[i * DMODEL + col];
    hh[r][col] = fmaxf(a2, 0.f);
    __syncthreads();

    if (col == 0) {
        float iu = ib2[0];
        for (int i = 0; i < DMODEL; ++i) iu += hh[r][i] * iw2[i];
        utilFinal[row] = utilBase[row] + attn_scale[0] * iu;
    }
}

// ---------------------------------------------------------------------------
// Kernel 5: per-batch outside option + softmax over N+1 alternatives.
// ---------------------------------------------------------------------------
__global__ __launch_bounds__(256) void final_kernel(
    const float* __restrict__ X,
    const float* __restrict__ outw1, const float* __restrict__ outb1,
    const float* __restrict__ outw2, const float* __restrict__ outb2,
    const float* __restrict__ outside_bias,
    const float* __restrict__ utilFinal,
    float* __restrict__ out)
{
    __shared__ float ctx[NPF];
    __shared__ float hid[32];
    __shared__ float red[256];
    __shared__ float u0s, gms, tots;
    const int b = blockIdx.x, t = threadIdx.x;

    if (t < NPF) {
        float s = 0.f;
        for (int n = 0; n < NTOK; ++n)
            s += X[((size_t)b * NTOK + n) * 32 + 1 + t];
        ctx[t] = s * (1.f / NTOK);
    }
    __syncthreads();
    if (t < 32) {
        float a = outb1[t];
        for (int i = 0; i < NPF; ++i) a += ctx[i] * outw1[i * 32 + t];
        hid[t] = fmaxf(a, 0.f);
    }
    __syncthreads();
    if (t == 0) {
        float u = outb2[0] + outside_bias[0];
        for (int i = 0; i < 32; ++i) u += hid[i] * outw2[i];
        u0s = u;
    }
    __syncthreads();

    const float* ub = utilFinal + (size_t)b * NTOK;
    float mx = u0s;
    for (int i = t; i < NTOK; i += 256) mx = fmaxf(mx, ub[i]);
    red[t] = mx; __syncthreads();
    for (int s = 128; s > 0; s >>= 1) {
        if (t < s) red[t] = fmaxf(red[t], red[t + s]);
        __syncthreads();
    }
    if (t == 0) gms = red[0];
    __syncthreads();
    const float gm = gms;

    float sm = 0.f;
    for (int i = t; i < NTOK; i += 256) sm += __expf(ub[i] - gm);
    red[t] = sm; __syncthreads();
    for (int s = 128; s > 0; s >>= 1) {
        if (t < s) red[t] += red[t + s];
        __syncthreads();
    }
    if (t == 0) tots = red[0] + __expf(u0s - gm);
    __syncthreads();

    const float inv = 1.f / tots;
    for (int i = t; i < NTOK; i += 256)
        out[(size_t)b * (NTOK + 1) + i] = __expf(ub[i] - gm) * inv;
    if (t == 0)
        out[(size_t)b * (NTOK + 1) + NTOK] = __expf(u0s - gm) * inv;
}

// ---------------------------------------------------------------------------
extern "C" void kernel_launch(void* const* d_in, const int* in_sizes, int n_in,
                              void* d_out, int out_size, void* d_ws, size_t ws_size,
                              hipStream_t stream) {
    const float* X        = (const float*)d_in[0];
    const float* sim      = (const float*)d_in[1];
    const float* bw1      = (const float*)d_in[2];
    const float* bb1      = (const float*)d_in[3];
    const float* bw2      = (const float*)d_in[4];
    const float* bb2      = (const float*)d_in[5];
    const float* bw3      = (const float*)d_in[6];
    const float* bb3      = (const float*)d_in[7];
    const float* w_raw    = (const float*)d_in[8];
    const float* a_raw    = (const float*)d_in[9];
    const float* cs       = (const float*)d_in[10];
    const float* p_bias   = (const float*)d_in[11];
    const float* fpw      = (const float*)d_in[12];
    const float* fpb      = (const float*)d_in[13];
    const float* qw       = (const float*)d_in[14];
    const float* qb       = (const float*)d_in[15];
    const float* kw       = (const float*)d_in[16];
    const float* kb       = (const float*)d_in[17];
    const float* vw       = (const float*)d_in[18];
    const float* vb       = (const float*)d_in[19];
    const float* ow       = (const float*)d_in[20];
    const float* ob       = (const float*)d_in[21];
    const float* tau_raw  = (const float*)d_in[22];
    const float* iw1      = (const float*)d_in[23];
    const float* ib1      = (const float*)d_in[24];
    const float* iw2      = (const float*)d_in[25];
    const float* ib2      = (const float*)d_in[26];
    const float* outw1    = (const float*)d_in[27];
    const float* outb1    = (const float*)d_in[28];
    const float* outw2    = (const float*)d_in[29];
    const float* outb2    = (const float*)d_in[30];
    const float* outside  = (const float*)d_in[31];
    const float* ascale   = (const float*)d_in[32];

    float* ws = (float*)d_ws;
    const size_t QKV = (size_t)BATCH * HEADS * NTOK * DHEAD; // 524288
    float* Q         = ws;
    float* Kt        = ws + QKV;
    float* V         = ws + 2 * QKV;
    float* AO        = ws + 3 * QKV;
    float* biasD     = ws + 4 * QKV;                          // N*N floats
    float* utilBase  = ws + 4 * QKV + (size_t)NTOK * NTOK;
    float* utilFinal = utilBase + BATCH * NTOK;

    prep_kernel<<<(BATCH * NTOK) / 4, 256, 0, stream>>>(
        X, bw1, bb1, bw2, bb2, bw3, bb3, w_raw, a_raw, cs, p_bias,
        fpw, fpb, qw, qb, kw, kb, vw, vb, Q, Kt, V, utilBase);

    topk_kernel<<<NTOK / 8, 256, 0, stream>>>(sim, tau_raw, biasD);

    flash_kernel<<<(BATCH * HEADS * (NTOK / 16)) / 8, 256, 0, stream>>>(
        Q, Kt, V, biasD, AO);

    post_kernel<<<(BATCH * NTOK) / 4, 256, 0, stream>>>(
        AO, ow, ob, iw1, ib1, iw2, ib2, utilBase, ascale, utilFinal);

    final_kernel<<<BATCH, 256, 0, stream>>>(
        X, outw1, outb1, outw2, outb2, outside, utilFinal, (float*)d_out);
}